// MHC_91122026152101
// MI455X (gfx1250) — compile-verified
//
#include <hip/hip_runtime.h>
#include <math.h>

typedef float v2f __attribute__((ext_vector_type(2)));
typedef float v8f __attribute__((ext_vector_type(8)));
typedef unsigned int v4u __attribute__((ext_vector_type(4)));
typedef int v4i __attribute__((ext_vector_type(4)));
typedef int v8i __attribute__((ext_vector_type(8)));

#define RMS_EPS   1e-6f
#define SINK_EPS  1e-12f
#define N_ITER    5

#if __has_builtin(__builtin_amdgcn_tensor_load_to_lds) && \
    __has_builtin(__builtin_amdgcn_s_wait_tensorcnt)
#define USE_TDM 1
#else
#define USE_TDM 0
#endif

// =====================================================================
// Kernel 1: per-token RMS sum-of-squares + 24 gate dot products,
// then sigmoid / 2*sigmoid / 4x4 sinkhorn finalization in lane 0.
// =====================================================================
__global__ __launch_bounds__(256) void gates_kernel(
    const float* __restrict__ x,        // (tokens, FD)
    const float* __restrict__ norm_w,   // (FD)
    const float* __restrict__ w_pre,    // (4, FD)
    const float* __restrict__ w_res,    // (16, FD)
    const float* __restrict__ w_post,   // (4, FD)
    const float* __restrict__ alpha_pre,
    const float* __restrict__ alpha_res,
    const float* __restrict__ alpha_post,
    const float* __restrict__ s_pre,    // (4)
    const float* __restrict__ s_res,    // (16)
    const float* __restrict__ s_post,   // (4)
    float* __restrict__ g_a,            // (tokens, 4)
    float* __restrict__ g_c,            // (tokens, 4)
    float* __restrict__ g_b,            // (tokens, 16)
    int FD)
{
    __shared__ float red[25 * 256];
    const int tok = blockIdx.x;
    const int tid = threadIdx.x;
    const float* flat = x + (size_t)tok * FD;

    float acc[25];
#pragma unroll
    for (int i = 0; i < 25; ++i) acc[i] = 0.0f;

    for (int k = tid; k < FD; k += 256) {
        const float xv = flat[k];
        const float v  = xv * norm_w[k];
        acc[24] += xv * xv;
        const float* wp = w_pre  + k;
        const float* wr = w_res  + k;
        const float* wq = w_post + k;
#pragma unroll
        for (int j = 0; j < 4; ++j)  acc[j]      += v * wp[(size_t)j * FD];
#pragma unroll
        for (int j = 0; j < 16; ++j) acc[4 + j]  += v * wr[(size_t)j * FD];
#pragma unroll
        for (int j = 0; j < 4; ++j)  acc[20 + j] += v * wq[(size_t)j * FD];
    }

#pragma unroll
    for (int i = 0; i < 25; ++i) red[i * 256 + tid] = acc[i];
    __syncthreads();
    for (int s = 128; s > 0; s >>= 1) {
        if (tid < s) {
#pragma unroll
            for (int i = 0; i < 25; ++i)
                red[i * 256 + tid] += red[i * 256 + tid + s];
        }
        __syncthreads();
    }

    if (tid == 0) {
        const float ssq = red[24 * 256];
        const float r   = rsqrtf(ssq / (float)FD + RMS_EPS);
        const float ap = alpha_pre[0], ar = alpha_res[0], ac = alpha_post[0];
#pragma unroll
        for (int h = 0; h < 4; ++h) {
            const float a_raw = ap * red[h * 256] * r + s_pre[h];
            g_a[tok * 4 + h]  = 1.0f / (1.0f + __expf(-a_raw));
            const float c_raw = ac * red[(20 + h) * 256] * r + s_post[h];
            g_c[tok * 4 + h]  = 2.0f / (1.0f + __expf(-c_raw));
        }
        float m[16];
        float mx = -1e30f;
#pragma unroll
        for (int i = 0; i < 16; ++i) {
            m[i] = ar * red[(4 + i) * 256] * r + s_res[i];
            mx = fmaxf(mx, m[i]);
        }
#pragma unroll
        for (int i = 0; i < 16; ++i) m[i] = __expf(m[i] - mx);
        for (int it = 0; it < N_ITER; ++it) {
#pragma unroll
            for (int row = 0; row < 4; ++row) {
                const float s = m[row*4+0] + m[row*4+1] + m[row*4+2] + m[row*4+3] + SINK_EPS;
                const float inv = 1.0f / s;
                m[row*4+0] *= inv; m[row*4+1] *= inv; m[row*4+2] *= inv; m[row*4+3] *= inv;
            }
#pragma unroll
            for (int col = 0; col < 4; ++col) {
                const float s = m[col] + m[4+col] + m[8+col] + m[12+col] + SINK_EPS;
                const float inv = 1.0f / s;
                m[col] *= inv; m[4+col] *= inv; m[8+col] *= inv; m[12+col] *= inv;
            }
        }
#pragma unroll
        for (int i = 0; i < 16; ++i) g_b[tok * 16 + i] = m[i];
    }
}

// =====================================================================
// Kernel 2: inner_in[tok,d] = sum_h a_l[tok,h] * x[tok,h,d]
// =====================================================================
__global__ __launch_bounds__(256) void inner_in_kernel(
    const float* __restrict__ x, const float* __restrict__ g_a,
    float* __restrict__ inner_in, int D)
{
    const int idx = blockIdx.x * 256 + threadIdx.x;  // tok*D + d
    const int tok = idx / D;
    const int d   = idx - tok * D;
    const float* xt = x + ((size_t)tok * 4) * D + d;
    const float a0 = g_a[tok*4+0], a1 = g_a[tok*4+1];
    const float a2 = g_a[tok*4+2], a3 = g_a[tok*4+3];
    inner_in[idx] = a0 * xt[0] + a1 * xt[(size_t)D]
                  + a2 * xt[(size_t)2*D] + a3 * xt[(size_t)3*D];
}

// =====================================================================
// Kernel 3: fp32 WMMA GEMM  C[M,N] = A[M,K] * Bw[N,K]^T
//   WG tile 128M x 128N, K chunk 32, double-buffered LDS (64KB).
//   8 waves as 4(M) x 2(N); wave tile 32M x 64N = 8 v8f accumulators.
//   Staging: Tensor Data Mover (tensor_load_to_lds + s_wait_tensorcnt)
//   pipelined one chunk ahead; fallback = vector-load staging.
// =====================================================================
#define K_CHUNK 32
#define WG_M    128
#define WG_N    128

#if USE_TDM
// Issue one TDM 2D tile load: tileRows x 32 floats, row stride = rowLen.
__device__ __forceinline__ void tdm_tile_load(
    const float* gsrc,          // start of tile in global memory
    unsigned ldsByteAddr,       // LDS destination byte address
    unsigned rowLen,            // tensor_dim0 & tensor_dim0_stride (elements)
    unsigned nRows,             // tensor_dim1 (elements)
    unsigned tileRows)          // tile_dim1
{
    const unsigned long long gaddr = (unsigned long long)(uintptr_t)gsrc;
    // ---- D# group 0: count=1, lds_addr, 57-bit global addr, type=2
    v4u g0 = { 1u,
               ldsByteAddr,
               (unsigned)(gaddr & 0xFFFFFFFFull),
               (unsigned)((gaddr >> 32) & 0x1FFFFFFull) | (2u << 30) };
    // ---- D# group 1: data_size=4B (code 2), dims/strides, tile dims
    const unsigned d0 = rowLen, d1 = nRows;
    const unsigned long long s0 = rowLen;           // dim0 stride (elements)
    v8i g1 = { (int)0x00020000u,                               // mask=0, data_size=2
               (int)((d0 & 0xFFFFu) << 16),                    // dim0[15:0] @ [63:48]
               (int)((d0 >> 16) | ((d1 & 0xFFFFu) << 16)),     // dim0[31:16], dim1[15:0]
               (int)((d1 >> 16) | ((unsigned)K_CHUNK << 16)),  // dim1[31:16], tile_dim0
               (int)(tileRows & 0xFFFFu),                      // tile_dim1, tile_dim2=0
               (int)(s0 & 0xFFFFFFFFull),                      // dim0_stride[31:0]
               (int)((s0 >> 32) & 0xFFFFull),                  // dim0_stride[47:32]
               0 };
    v4i z4 = { 0, 0, 0, 0 };
#if __clang_major__ >= 23
    v8i z8 = { 0, 0, 0, 0, 0, 0, 0, 0 };
    __builtin_amdgcn_tensor_load_to_lds(g0, g1, z4, z4, z8, 0);
#else
    __builtin_amdgcn_tensor_load_to_lds(g0, g1, z4, z4, 0);
#endif
}
#endif // USE_TDM

__global__ __launch_bounds__(256) void gemm_wmma_f32(
    const float* __restrict__ A,   // (M, K) inner_in
    const float* __restrict__ Bw,  // (N, K) w_inner
    float* __restrict__ C,         // (M, N) inner_out
    int M, int N, int K)
{
    __shared__ float lds_a[2][WG_M * K_CHUNK];   // 2 x 16 KB
    __shared__ float lds_b[2][WG_N * K_CHUNK];   // 2 x 16 KB

    const int tid   = threadIdx.x;
    const int lane  = tid & 31;
    const int wave  = tid >> 5;        // 0..7
    const int waveM = wave >> 1;       // 0..3 : 32-row M subtile
    const int waveN = wave & 1;        // 0..1 : 64-col N subtile
    const int half  = lane >> 4;       // lane half: K pair select / M+8
    const int l16   = lane & 15;

    const int m0 = blockIdx.y * WG_M;
    const int n0 = blockIdx.x * WG_N;
    const int numChunks = K / K_CHUNK;

    v8f acc[2][4] = {};

#if USE_TDM
    // LDS byte offsets of the staging buffers (generic-pointer low 32 bits).
    const unsigned ldsA0 = (unsigned)(uintptr_t)&lds_a[0][0];
    const unsigned ldsA1 = (unsigned)(uintptr_t)&lds_a[1][0];
    const unsigned ldsB0 = (unsigned)(uintptr_t)&lds_b[0][0];
    const unsigned ldsB1 = (unsigned)(uintptr_t)&lds_b[1][0];

    // Prologue: DMA chunk 0 into buffer 0 (one wave drives the TDM).
    if (wave == 0) {
        tdm_tile_load(&A [(size_t)m0 * K], ldsA0, (unsigned)K, (unsigned)M, WG_M);
        tdm_tile_load(&Bw[(size_t)n0 * K], ldsB0, (unsigned)K, (unsigned)N, WG_N);
    }
#endif

    for (int c = 0; c < numChunks; ++c) {
        const int buf = c & 1;

#if USE_TDM
        if (wave == 0) {
            if (c + 1 < numChunks) {
                const int k1 = (c + 1) * K_CHUNK;
                tdm_tile_load(&A [(size_t)m0 * K + k1], buf ? ldsA0 : ldsA1,
                              (unsigned)K, (unsigned)M, WG_M);
                tdm_tile_load(&Bw[(size_t)n0 * K + k1], buf ? ldsB0 : ldsB1,
                              (unsigned)K, (unsigned)N, WG_N);
                // In-order TDM completion: <=2 outstanding => chunk c landed.
                __builtin_amdgcn_s_wait_tensorcnt(2);
            } else {
                __builtin_amdgcn_s_wait_tensorcnt(0);
            }
        }
        __syncthreads();               // chunk c visible to all waves
#else
        // Fallback staging: coalesced float4 copies.
        const int k0 = c * K_CHUNK;
#pragma unroll
        for (int i = 0; i < 4; ++i) {  // A: 1024 float4
            const int vid = tid + 256 * i;
            const int row = vid >> 3;
            const int kq  = (vid & 7) * 4;
            *(float4*)&lds_a[buf][row * K_CHUNK + kq] =
                *(const float4*)&A[(size_t)(m0 + row) * K + k0 + kq];
        }
#pragma unroll
        for (int i = 0; i < 4; ++i) {  // B: 1024 float4
            const int vid = tid + 256 * i;
            const int row = vid >> 3;
            const int kq  = (vid & 7) * 4;
            *(float4*)&lds_b[buf][row * K_CHUNK + kq] =
                *(const float4*)&Bw[(size_t)(n0 + row) * K + k0 + kq];
        }
        __syncthreads();
#endif

        // ---- compute: per K-substep, 2 A-frags + 4 B-frags feed 8 WMMAs
        const float* aBase0 = &lds_a[buf][(waveM * 32 +  0 + l16) * K_CHUNK + 2 * half];
        const float* aBase1 = &lds_a[buf][(waveM * 32 + 16 + l16) * K_CHUNK + 2 * half];
        const float* bBase  = &lds_b[buf][(waveN * 64 +      l16) * K_CHUNK + 2 * half];
#pragma unroll
        for (int kk = 0; kk < K_CHUNK; kk += 4) {
            const v2f a0 = *(const v2f*)(aBase0 + kk);
            const v2f a1 = *(const v2f*)(aBase1 + kk);
            v2f b[4];
#pragma unroll
            for (int ns = 0; ns < 4; ++ns)
                b[ns] = *(const v2f*)(bBase + (size_t)ns * 16 * K_CHUNK + kk);
#pragma unroll
            for (int ns = 0; ns < 4; ++ns)
                acc[0][ns] = __builtin_amdgcn_wmma_f32_16x16x4_f32(
                    false, a0, false, b[ns], (short)0, acc[0][ns], false, false);
#pragma unroll
            for (int ns = 0; ns < 4; ++ns)
                acc[1][ns] = __builtin_amdgcn_wmma_f32_16x16x4_f32(
                    false, a1, false, b[ns], (short)0, acc[1][ns], false, false);
        }
        __syncthreads();               // all reads of buf done before reuse
    }

    // ---- store: D layout — VGPR e = rows M=e (lanes 0-15) / M=e+8 (16-31)
#pragma unroll
    for (int ms = 0; ms < 2; ++ms) {
#pragma unroll
        for (int ns = 0; ns < 4; ++ns) {
#pragma unroll
            for (int e = 0; e < 8; ++e) {
                const int m = m0 + waveM * 32 + ms * 16 + e + 8 * half;
                const int n = n0 + waveN * 64 + ns * 16 + l16;
                C[(size_t)m * N + n] = acc[ms][ns][e];
            }
        }
    }
}

// =====================================================================
// Kernel 4: out[tok,h,d] = sum_k b_l[tok,h,k]*x[tok,k,d] + c_l[tok,h]*inner_out[tok,d]
// =====================================================================
__global__ __launch_bounds__(256) void combine_kernel(
    const float* __restrict__ x, const float* __restrict__ g_b,
    const float* __restrict__ g_c, const float* __restrict__ inner_out,
    float* __restrict__ out, int D)
{
    const int idx = blockIdx.x * 256 + threadIdx.x;  // tok*D + d
    const int tok = idx / D;
    const int d   = idx - tok * D;
    const float* xt = x + ((size_t)tok * 4) * D + d;
    const float x0 = xt[0];
    const float x1 = xt[(size_t)D];
    const float x2 = xt[(size_t)2*D];
    const float x3 = xt[(size_t)3*D];
    const float io = inner_out[idx];
    const float* bl = g_b + tok * 16;
    const float* cl = g_c + tok * 4;
    float* ot = out + ((size_t)tok * 4) * D + d;
#pragma unroll
    for (int h = 0; h < 4; ++h) {
        ot[(size_t)h * D] = bl[h*4+0]*x0 + bl[h*4+1]*x1 + bl[h*4+2]*x2
                          + bl[h*4+3]*x3 + cl[h]*io;
    }
}

// =====================================================================
extern "C" void kernel_launch(void* const* d_in, const int* in_sizes, int n_in,
                              void* d_out, int out_size, void* d_ws, size_t ws_size,
                              hipStream_t stream) {
    const float* x          = (const float*)d_in[0];
    const float* norm_w     = (const float*)d_in[1];
    const float* w_pre      = (const float*)d_in[2];
    const float* w_res      = (const float*)d_in[3];
    const float* w_post     = (const float*)d_in[4];
    const float* alpha_pre  = (const float*)d_in[5];
    const float* alpha_res  = (const float*)d_in[6];
    const float* alpha_post = (const float*)d_in[7];
    const float* s_pre      = (const float*)d_in[8];
    const float* s_res      = (const float*)d_in[9];
    const float* s_post     = (const float*)d_in[10];
    const float* w_inner    = (const float*)d_in[11];
    float* out = (float*)d_out;

    const int FD     = in_sizes[1];          // 8192
    const int H      = in_sizes[8];          // 4
    const int D      = FD / H;               // 2048
    const int tokens = in_sizes[0] / FD;     // 4096 = B*T

    float* ws        = (float*)d_ws;
    float* g_a       = ws;
    float* g_c       = g_a + (size_t)tokens * 4;
    float* g_b       = g_c + (size_t)tokens * 4;
    float* inner_in  = g_b + (size_t)tokens * 16;
    float* inner_out = inner_in + (size_t)tokens * D;

    gates_kernel<<<tokens, 256, 0, stream>>>(
        x, norm_w, w_pre, w_res, w_post,
        alpha_pre, alpha_res, alpha_post,
        s_pre, s_res, s_post,
        g_a, g_c, g_b, FD);

    const int nElem = tokens * D;
    inner_in_kernel<<<nElem / 256, 256, 0, stream>>>(x, g_a, inner_in, D);

    dim3 grid(D / WG_N, tokens / WG_M);      // (16, 32)
    gemm_wmma_f32<<<grid, 256, 0, stream>>>(inner_in, w_inner, inner_out,
                                            tokens, D, D);

    combine_kernel<<<nElem / 256, 256, 0, stream>>>(x, g_b, g_c, inner_out,
                                                    out, D);
}